// LocalAggregation_11252814315649
// MI455X (gfx1250) — compile-verified
//
#include <hip/hip_runtime.h>

// ---------------- problem constants (from reference) ----------------
#define NPTS   8192
#define BATCH  2
#define KNB    32          // nsample
#define CFEAT  64          // input feature channels
#define CIN    67          // 3 + 64 (dp_fj)
#define CINP   68          // padded to multiple of 4 (even -> 8B-aligned v2f)
#define C1     128
#define C2     128
#define RADIUS2 (0.15f * 0.15f)
#define BN_EPS  1e-5f

// Permuted input-channel layout (W1p columns permuted identically):
//   c' = 0..63  -> fj channel c'      (16B-aligned float4 rows in LDS)
//   c' = 64..66 -> dp dims 0..2
//   c' = 67     -> zero pad
#define XS_STRIDE 68       // Xs[k][c']  neighbor-major
#define HS_STRIDE 130      // Hs[k][ch]  neighbor-major, padded (even, !=0 mod 64)

#define GRID_FUSED 2048    // persistent-ish: 16384 points / 2048 = 8 per block

#define IDX_BYTES  ((size_t)BATCH * NPTS * KNB * sizeof(int))   // 2 MB
#define W1P_BYTES  ((size_t)C1 * CINP * sizeof(float))          // 34816 B
#define COEF_BYTES ((size_t)4 * C1 * sizeof(float))             // 2048 B
#define FT_BYTES   ((size_t)BATCH * NPTS * CFEAT * sizeof(float)) // 4 MB

typedef __attribute__((ext_vector_type(2))) float v2f;
typedef __attribute__((ext_vector_type(8))) float v8f;

// =====================================================================
// Kernel 0a: one-time prep of weights + BN coefficients.
//  - W1p [128][68]: column-permuted (fj first, dp at 64..66, pad 67) so
//    A-fragments are guard-free aligned v2f loads matching the Xs layout.
//  - coef: interleaved {scale, shift}; [0..255] layer1, [256..511] layer2.
// =====================================================================
__global__ __launch_bounds__(256) void prep_kernel(
    const float* __restrict__ W1,
    const float* __restrict__ gamma1, const float* __restrict__ beta1,
    const float* __restrict__ mean1,  const float* __restrict__ var1,
    const float* __restrict__ gamma2, const float* __restrict__ beta2,
    const float* __restrict__ mean2,  const float* __restrict__ var2,
    float* __restrict__ W1p, float* __restrict__ coef)
{
    const int e = blockIdx.x * 256 + threadIdx.x;
    if (e < C1 * CINP) {
        const int m  = e / CINP;
        const int cp = e % CINP;
        float w;
        if (cp < CFEAT)        w = W1[m * CIN + 3 + cp];   // fj columns
        else if (cp < CIN)     w = W1[m * CIN + (cp - CFEAT)]; // dp columns
        else                   w = 0.0f;                   // pad
        W1p[e] = w;
    }
    if (e < C1) {
        const float sc1 = gamma1[e] * rsqrtf(var1[e] + BN_EPS);
        coef[2 * e]     = sc1;
        coef[2 * e + 1] = beta1[e] - mean1[e] * sc1;
        const float sc2 = gamma2[e] * rsqrtf(var2[e] + BN_EPS);
        coef[2 * C1 + 2 * e]     = sc2;
        coef[2 * C1 + 2 * e + 1] = beta2[e] - mean2[e] * sc2;
    }
}

// =====================================================================
// Kernel 0b: transpose f [B,64,N] -> f_t [B,N,64] (one-time 4 MB) so a
// neighbor's 64 channels are one contiguous 256B row (float4-gatherable).
// =====================================================================
__global__ __launch_bounds__(256) void transpose_f_kernel(
    const float* __restrict__ f, float* __restrict__ f_t)
{
    const int e = blockIdx.x * 256 + threadIdx.x;   // (b, c, n), n fastest
    if (e >= BATCH * CFEAT * NPTS) return;
    const int n = e % NPTS;
    const int c = (e / NPTS) % CFEAT;
    const int b = e / (NPTS * CFEAT);
    f_t[((size_t)b * NPTS + n) * CFEAT + c] = f[e];  // coalesced reads
}

// =====================================================================
// Kernel 1: ball query. One thread per query point; LDS-tiled scan of
// all points in increasing j, collecting the first KNB hits (matches
// the reference's sort-then-slice semantics). Pads with the first hit.
// =====================================================================
__global__ __launch_bounds__(256) void ball_query_kernel(
    const float* __restrict__ p, int* __restrict__ idx_out)
{
    __shared__ float sx[256], sy[256], sz[256];

    const int q = blockIdx.x * 256 + threadIdx.x;   // 0 .. B*N-1
    const int b = q / NPTS;                         // uniform per block
    const int n = q % NPTS;

    const float qx = p[(b * NPTS + n) * 3 + 0];
    const float qy = p[(b * NPTS + n) * 3 + 1];
    const float qz = p[(b * NPTS + n) * 3 + 2];

    int cnt = 0;
    int firstj = n;
    int* myout = idx_out + q * KNB;

    for (int t = 0; t < NPTS / 256; ++t) {
        const int j0 = t * 256;
        __syncthreads();
        {
            const float* pj = p + (b * NPTS + j0 + threadIdx.x) * 3;
            sx[threadIdx.x] = pj[0];
            sy[threadIdx.x] = pj[1];
            sz[threadIdx.x] = pj[2];
        }
        __syncthreads();
        if (cnt < KNB) {
            for (int jj = 0; jj < 256 && cnt < KNB; ++jj) {
                const float dx = sx[jj] - qx;
                const float dy = sy[jj] - qy;
                const float dz = sz[jj] - qz;
                const float d2 = dx * dx + dy * dy + dz * dz;
                if (d2 < RADIUS2) {
                    if (cnt == 0) firstj = j0 + jj;
                    myout[cnt++] = j0 + jj;
                }
            }
        }
    }
    for (int c = cnt; c < KNB; ++c) myout[c] = firstj;   // PointNet++ padding
}

// =====================================================================
// Kernel 2: fused gather + conv1(BN,ReLU) + conv2(BN,ReLU) + max-over-K.
// Persistent 256-thread blocks (8 wave32); each block loops over 8 points.
// Point-invariant A-fragments (W1p/W2 rows of this wave's m-tile) are
// preloaded ONCE into registers (v2f a1[17], a2[32]); the per-point inner
// loops are pure ds_load_b64 + v_wmma_f32_16x16x4_f32.
//
// LDS is NEIGHBOR-MAJOR: each lane's B fragment is one aligned ds_load_b64.
//   Xs[k * 68 + c']  banks (4n+k0)%64 over lanes -> conflict-free
//   Hs[k * 130 + ch] banks (2n+k0)%64 over lanes -> conflict-free
//
// WMMA f32 16x16x4 fragment layouts (ISA 7.12.2), wave32:
//   A (16x4, MxK):  lane = m + 16*(k>>1), vgpr = k&1
//   B (4x16, KxN):  lane = n + 16*(k>>1), vgpr = k&1
//   C/D (16x16):    vgpr r, lanes 0-15 -> (M=r, N=lane),
//                            lanes 16-31 -> (M=r+8, N=lane-16)
// =====================================================================
__global__ __launch_bounds__(256) void fused_pointconv_kernel(
    const float* __restrict__ p, const float* __restrict__ f_t,
    const int* __restrict__ idx, const float* __restrict__ W1p,
    const float* __restrict__ W2, const float* __restrict__ coef,
    float* __restrict__ out)
{
    __shared__ float Xs[KNB * XS_STRIDE];   // [32][68]  neighbor-major
    __shared__ float Hs[KNB * HS_STRIDE];   // [32][130] neighbor-major
    __shared__ int   Is[KNB];

    const int tid  = threadIdx.x;
    const int lane = tid & 31;
    const int wave = tid >> 5;       // 0..7 -> output-channel tile (m-tile)
    const int half = lane >> 4;      // 0/1: K sub-half of the fragment
    const int lm   = lane & 15;
    const int m    = wave * 16 + lm; // A-fragment row

    // ---- preload point-invariant weight fragments into registers ----
    v2f a1[CINP / 4];                // 17 x v2f = 34 VGPRs
    v2f a2[C1 / 4];                  // 32 x v2f = 64 VGPRs
    #pragma unroll
    for (int kk = 0; kk < CINP / 4; ++kk)
        a1[kk] = *(const v2f*)&W1p[m * CINP + kk * 4 + 2 * half];
    #pragma unroll
    for (int kk = 0; kk < C1 / 4; ++kk)
        a2[kk] = *(const v2f*)&W2[m * C1 + kk * 4 + 2 * half];

    for (int bn = blockIdx.x; bn < BATCH * NPTS; bn += GRID_FUSED) {
        const int b = bn / NPTS;
        const int n = bn % NPTS;

        if (tid < KNB) Is[tid] = idx[bn * KNB + tid];
        __syncthreads();

        // ---- gather fj rows as float4 (16B-aligned in Xs, c'=0..63) ----
        for (int e4 = tid; e4 < KNB * (CFEAT / 4); e4 += 256) {  // 2 iters
            const int k  = e4 >> 4;
            const int q4 = e4 & 15;
            const int j  = Is[k];
            const float4 v = *(const float4*)&f_t[((size_t)b * NPTS + j) * CFEAT + 4 * q4];
            *(float4*)&Xs[k * XS_STRIDE + 4 * q4] = v;
        }
        // ---- dp at c'=64..66, pad at c'=67 ----
        if (tid < KNB * 4) {
            const int k = tid >> 2;
            const int c = tid & 3;
            float val = 0.0f;
            if (c < 3) {
                const int j = Is[k];
                val = p[(b * NPTS + j) * 3 + c] - p[(b * NPTS + n) * 3 + c];
            }
            Xs[k * XS_STRIDE + CFEAT + c] = val;
        }
        __syncthreads();

        // ---------------- conv1: (128x68) x (68x32) ----------------
        v8f acc0 = {};               // neighbor cols 0..15
        v8f acc1 = {};               // neighbor cols 16..31
        #pragma unroll
        for (int kk = 0; kk < CINP / 4; ++kk) {
            const int k0 = kk * 4 + 2 * half;
            const v2f b0 = *(const v2f*)&Xs[lm        * XS_STRIDE + k0];
            const v2f b1 = *(const v2f*)&Xs[(lm + 16) * XS_STRIDE + k0];
            acc0 = __builtin_amdgcn_wmma_f32_16x16x4_f32(false, a1[kk], false, b0,
                                                         (short)0, acc0, false, false);
            acc1 = __builtin_amdgcn_wmma_f32_16x16x4_f32(false, a1[kk], false, b1,
                                                         (short)0, acc1, false, false);
        }

        // ---- BN1 + ReLU (precomputed {scale,shift}), scatter Hs[k][ch] ----
        #pragma unroll
        for (int r = 0; r < 8; ++r) {
            const int ch = wave * 16 + 8 * half + r;
            const v2f ss = *(const v2f*)&coef[2 * ch];
            Hs[lm        * HS_STRIDE + ch] = fmaxf(acc0[r] * ss.x + ss.y, 0.0f);
            Hs[(lm + 16) * HS_STRIDE + ch] = fmaxf(acc1[r] * ss.x + ss.y, 0.0f);
        }
        __syncthreads();

        // ---------------- conv2: (128x128) x (128x32) ----------------
        v8f d0 = {};
        v8f d1 = {};
        #pragma unroll
        for (int kk = 0; kk < C1 / 4; ++kk) {
            const int k0 = kk * 4 + 2 * half;
            const v2f b0 = *(const v2f*)&Hs[lm        * HS_STRIDE + k0];
            const v2f b1 = *(const v2f*)&Hs[(lm + 16) * HS_STRIDE + k0];
            d0 = __builtin_amdgcn_wmma_f32_16x16x4_f32(false, a2[kk], false, b0,
                                                       (short)0, d0, false, false);
            d1 = __builtin_amdgcn_wmma_f32_16x16x4_f32(false, a2[kk], false, b1,
                                                       (short)0, d1, false, false);
        }

        // ---- BN2 + ReLU + max over 32 neighbors ----
        #pragma unroll
        for (int r = 0; r < 8; ++r) {
            const int ch = wave * 16 + 8 * half + r;
            const v2f ss = *(const v2f*)&coef[2 * C1 + 2 * ch];
            float mx = fmaxf(fmaxf(d0[r] * ss.x + ss.y, 0.0f),
                             fmaxf(d1[r] * ss.x + ss.y, 0.0f));
            #pragma unroll
            for (int s = 1; s < 16; s <<= 1)
                mx = fmaxf(mx, __shfl_xor(mx, s, 32));
            if (lm == 0) out[(b * C2 + ch) * NPTS + n] = mx;
        }
        __syncthreads();   // protect Is/Xs/Hs reuse for next point
    }
}

// =====================================================================
extern "C" void kernel_launch(void* const* d_in, const int* in_sizes, int n_in,
                              void* d_out, int out_size, void* d_ws, size_t ws_size,
                              hipStream_t stream)
{
    const float* p      = (const float*)d_in[0];
    const float* f      = (const float*)d_in[1];
    const float* W1     = (const float*)d_in[2];
    const float* gamma1 = (const float*)d_in[3];
    const float* beta1  = (const float*)d_in[4];
    const float* mean1  = (const float*)d_in[5];
    const float* var1   = (const float*)d_in[6];
    const float* W2     = (const float*)d_in[7];
    const float* gamma2 = (const float*)d_in[8];
    const float* beta2  = (const float*)d_in[9];
    const float* mean2  = (const float*)d_in[10];
    const float* var2   = (const float*)d_in[11];
    float* out = (float*)d_out;

    char*  ws   = (char*)d_ws;
    int*   idx  = (int*)ws;                                       // 2 MB
    float* W1p  = (float*)(ws + IDX_BYTES);                       // 34.8 KB
    float* coef = (float*)(ws + IDX_BYTES + W1P_BYTES);           // 2 KB
    float* f_t  = (float*)(ws + IDX_BYTES + W1P_BYTES + COEF_BYTES); // 4 MB

    prep_kernel<<<(C1 * CINP + 255) / 256, 256, 0, stream>>>(
        W1, gamma1, beta1, mean1, var1, gamma2, beta2, mean2, var2, W1p, coef);

    transpose_f_kernel<<<(BATCH * CFEAT * NPTS + 255) / 256, 256, 0, stream>>>(f, f_t);

    ball_query_kernel<<<(BATCH * NPTS) / 256, 256, 0, stream>>>(p, idx);

    fused_pointconv_kernel<<<GRID_FUSED, 256, 0, stream>>>(
        p, f_t, idx, W1p, W2, coef, out);
}